// LightGCN_455266533420
// MI455X (gfx1250) — compile-verified
//
#include <hip/hip_runtime.h>
#include <hip/hip_bf16.h>
#include <stdint.h>

#define DFEAT 64
#define WAVE  32

// ---------------------------------------------------------------------------
// CDNA5 helpers
// ---------------------------------------------------------------------------

// Low 32 bits of a generic pointer into LDS are the workgroup-relative LDS
// byte offset (flat-aperture mapping, ISA 10.2): usable as the async-load
// LDS destination address.
__device__ __forceinline__ uint32_t lds_off(const void* p) {
  return (uint32_t)(uintptr_t)p;
}

// GVS-mode async copy: copies 32 bits from global address sbase + voff to the LDS address in lds; bumps ASYNCcnt
__device__ __forceinline__ void async_b32_to_lds(uint32_t lds, const void* sbase, int voff) {
  asm volatile("global_load_async_to_lds_b32 %0, %1, %2"
               :
               : "v"(lds), "v"(voff), "s"(sbase)
               : "memory");
}

template <int CNT>
__device__ __forceinline__ void wait_async() {
  asm volatile("s_wait_asynccnt %0" :: "n"(CNT) : "memory");
}

// ---------------------------------------------------------------------------
// SpMM: y[row[e], :] += val[e] * x[col[e], :]   (y pre-zeroed)
// One wave32 per 32-edge batch, grid-stride over batches. Edge triplets are
// staged global->LDS with async copies, double-buffered across batches
// (issue next batch, s_wait_asynccnt 3 -> current batch ready, 3 in flight).
// Per edge: v_readlane broadcast, float2/lane gather (L2-resident x),
// 2 no-return fp32 global atomics per lane into y[row].
// ---------------------------------------------------------------------------
__global__ __launch_bounds__(256)
void spmm_edges(const float* __restrict__ x,
                float* __restrict__ y,
                const int* __restrict__ row,
                const int* __restrict__ col,
                const float* __restrict__ val,
                int E) {
  __shared__ int   sRow[8][2][WAVE];
  __shared__ int   sCol[8][2][WAVE];
  __shared__ float sVal[8][2][WAVE];

  const int lane   = threadIdx.x & (WAVE - 1);
  const int w      = threadIdx.x >> 5;
  const int wavesB = blockDim.x >> 5;
  const int waveId = blockIdx.x * wavesB + w;
  const int nWaves = gridDim.x * wavesB;
  const int nBatch = (E + WAVE - 1) / WAVE;

  if (waveId >= nBatch) return;   // wave-uniform

  // Prologue: stage first batch into stage 0.
  {
    int e  = waveId * WAVE + lane;
    int ec = (e < E) ? e : (E - 1);
    async_b32_to_lds(lds_off(&sRow[w][0][lane]), row, ec * 4);
    async_b32_to_lds(lds_off(&sCol[w][0][lane]), col, ec * 4);
    async_b32_to_lds(lds_off(&sVal[w][0][lane]), val, ec * 4);
  }

  int stage = 0;
  for (int b = waveId; b < nBatch; b += nWaves) {
    const int bn = b + nWaves;
    if (bn < nBatch) {                       // wave-uniform branch
      // Stage the next batch into the alternate buffer, then wait for the
      // current batch only (ASYNCcnt completes in order).
      int e2  = bn * WAVE + lane;
      int ec2 = (e2 < E) ? e2 : (E - 1);
      const int s2 = stage ^ 1;
      async_b32_to_lds(lds_off(&sRow[w][s2][lane]), row, ec2 * 4);
      async_b32_to_lds(lds_off(&sCol[w][s2][lane]), col, ec2 * 4);
      async_b32_to_lds(lds_off(&sVal[w][s2][lane]), val, ec2 * 4);
      wait_async<3>();
    } else {
      wait_async<0>();
    }

    const int   r = sRow[w][stage][lane];
    const int   c = sCol[w][stage][lane];
    const float v = sVal[w][stage][lane];

    // Warm L2/L0 for this batch's 32 gather rows (global_prefetch_b8).
    const int base = b * WAVE;
    if (base + lane < E) __builtin_prefetch(x + (size_t)c * DFEAT, 0, 3);

    const int rem = E - base;
    const int cnt = (rem < WAVE) ? rem : WAVE;   // wave-uniform
    for (int k = 0; k < cnt; ++k) {
      const int   rk = __builtin_amdgcn_readlane(r, k);
      const int   ck = __builtin_amdgcn_readlane(c, k);
      const float vk = __int_as_float(__builtin_amdgcn_readlane(__float_as_int(v), k));

      const float2 xv = *((const float2*)(x + (size_t)ck * DFEAT) + lane);
      float* dst = y + (size_t)rk * DFEAT + lane * 2;
      unsafeAtomicAdd(dst,     vk * xv.x);   // global_atomic_add_f32 (no return)
      unsafeAtomicAdd(dst + 1, vk * xv.y);
    }
    stage ^= 1;
  }
}

// ---------------------------------------------------------------------------
// Fused elementwise kernels (float4 / B128 traffic)
// ---------------------------------------------------------------------------

// cur = out = concat(user,item); nxt = 0   (one pass)
__global__ __launch_bounds__(256)
void init_cat_zero(const float4* __restrict__ u, const float4* __restrict__ it,
                   float4* __restrict__ cur, float4* __restrict__ out,
                   float4* __restrict__ nxt, int nu4, int total4) {
  int i = blockIdx.x * blockDim.x + threadIdx.x;
  if (i >= total4) return;
  float4 v = (i < nu4) ? u[i] : it[i - nu4];
  cur[i] = v;
  out[i] = v;
  nxt[i] = make_float4(0.f, 0.f, 0.f, 0.f);
}

// out += nxt; zbuf = 0  (zbuf becomes the next layer's scatter target)
__global__ __launch_bounds__(256)
void acc_add_zero(float4* __restrict__ out, const float4* __restrict__ nx,
                  float4* __restrict__ zbuf, int n4) {
  int i = blockIdx.x * blockDim.x + threadIdx.x;
  if (i >= n4) return;
  float4 a = out[i], b = nx[i];
  a.x += b.x; a.y += b.y; a.z += b.z; a.w += b.w;
  out[i] = a;
  zbuf[i] = make_float4(0.f, 0.f, 0.f, 0.f);
}

// out = (out + nxt) * s   (final layer: fold in the 1/(L+1) scale)
__global__ __launch_bounds__(256)
void acc_add_scale(float4* __restrict__ out, const float4* __restrict__ nx,
                   float s, int n4) {
  int i = blockIdx.x * blockDim.x + threadIdx.x;
  if (i >= n4) return;
  float4 a = out[i], b = nx[i];
  a.x = (a.x + b.x) * s; a.y = (a.y + b.y) * s;
  a.z = (a.z + b.z) * s; a.w = (a.w + b.w) * s;
  out[i] = a;
}

// ---------------------------------------------------------------------------
// Launch: embed0 -> cur & out, nxt=0;
//   L-1 x { spmm(cur->nxt); out+=nxt, zero cur; swap }
//   last  { spmm(cur->nxt); out=(out+nxt)/(L+1) }
// ---------------------------------------------------------------------------
extern "C" void kernel_launch(void* const* d_in, const int* in_sizes, int n_in,
                              void* d_out, int out_size, void* d_ws, size_t ws_size,
                              hipStream_t stream) {
  const float* user = (const float*)d_in[0];
  const float* item = (const float*)d_in[1];
  const float* aval = (const float*)d_in[2];
  const int*   arow = (const int*)d_in[3];
  const int*   acol = (const int*)d_in[4];
  const int num_layers = 3;   // reference setup fixes num_layers = 3

  const int nu = in_sizes[0] / DFEAT;
  const int ni = in_sizes[1] / DFEAT;
  const int N  = nu + ni;
  const int E  = in_sizes[2];
  const size_t nd = (size_t)N * DFEAT;

  float* cur = (float*)d_ws;        // N*D floats
  float* nxt = cur + nd;            // N*D floats
  float* out = (float*)d_out;       // N*D floats (users || items)

  const int total4 = (int)(nd / 4);
  const int nu4    = nu * DFEAT / 4;

  dim3 blk(256);
  dim3 grdN((total4 + 255) / 256);
  dim3 grdE(4096);                  // 32768 waves over E/32 batches

  init_cat_zero<<<grdN, blk, 0, stream>>>((const float4*)user, (const float4*)item,
                                          (float4*)cur, (float4*)out, (float4*)nxt,
                                          nu4, total4);

  for (int l = 0; l < num_layers; ++l) {
    spmm_edges<<<grdE, blk, 0, stream>>>(cur, nxt, arow, acol, aval, E);
    if (l + 1 < num_layers) {
      // out += nxt; zero cur (it becomes the next scatter target after swap)
      acc_add_zero<<<grdN, blk, 0, stream>>>((float4*)out, (const float4*)nxt,
                                             (float4*)cur, total4);
      float* t = cur; cur = nxt; nxt = t;
    } else {
      acc_add_scale<<<grdN, blk, 0, stream>>>((float4*)out, (const float4*)nxt,
                                              1.0f / (float)(num_layers + 1), total4);
    }
  }
}